// TaskNearestAssigner_21646635172367
// MI455X (gfx1250) — compile-verified
//
#include <hip/hip_runtime.h>
#include <stdint.h>

// ---------------------------------------------------------------------------
// TaskNearestAssigner for MI455X (gfx1250, wave32)
//   b=32, n=64 GTs, 80x80 grid (ng=6400), NA=3 anchors, 80 classes, top-3.
// Store-bandwidth bound (~210MB out, > 192MB L2 -> NT stores); CDNA5 async
// global->LDS staging for per-batch metadata (ASYNCcnt path); LDS tree
// reductions sized for wave32.
// ---------------------------------------------------------------------------

#define B_    32
#define N_    64
#define GS_   80
#define NG_   6400
#define NA_   3
#define NC_   80
#define TOPK_ 3

// d_out layout (floats, concatenated in reference return order)
#define BB_CNT  (B_*NA_*NG_*4)        // 2,457,600  target_bboxes
#define SC_OFF  (BB_CNT)
#define SC_CNT  (B_*NA_*NG_*NC_)      // 49,152,000 target_scores (one-hot)
#define ANC_OFF (SC_OFF + SC_CNT)     // 51,609,600 anc (6 floats)
#define FG_OFF  (ANC_OFF + 6)         // fg_out (b,na,ng)

typedef float vfloat4 __attribute__((ext_vector_type(4)));

// ---- CDNA5 async global->LDS path (guarded; falls back to plain loads) ----
#if defined(__HIP_DEVICE_COMPILE__) && __has_builtin(__builtin_amdgcn_global_load_async_to_lds_b32)
#define HAVE_ASYNC_LDS 1
#else
#define HAVE_ASYNC_LDS 0
#endif

__device__ __forceinline__ void async_g2l_b32(const void* gsrc, void* ldst) {
#if HAVE_ASYNC_LDS
  // Builtin takes (as1 int* src, as3 int* dst, imm offset, imm cpol).
  // Global as(1) ptr has the same integer value as generic; LDS as(3) offset
  // is the low 32 bits of the generic shared pointer (ISA 10.2 aperture map).
  __builtin_amdgcn_global_load_async_to_lds_b32(
      (__attribute__((address_space(1))) int*)(uintptr_t)gsrc,
      (__attribute__((address_space(3))) int*)(uint32_t)(uintptr_t)ldst,
      0, 0);
#else
  *(uint32_t*)ldst = *(const uint32_t*)gsrc;
#endif
}

__device__ __forceinline__ void wait_async_lds() {
#if HAVE_ASYNC_LDS
#if __has_builtin(__builtin_amdgcn_s_wait_asynccnt)
  __builtin_amdgcn_s_wait_asynccnt(0);
#else
  asm volatile("s_wait_asynccnt 0" ::: "memory");
#endif
#endif
}

// ---------------------------------------------------------------------------
// Kernel A: per (b,n) find the 3 grid cells with smallest L1 distance.
// Key = (f32bits(dist) << 32) | cell_index  -> lexicographic (dist, idx)
// matches jax.lax.top_k tie-breaking (lower index wins). dist >= 0 so the
// float bit pattern is order-preserving as u32.
// Grid coords carried incrementally in exact small-int float arithmetic
// (g steps by 256 = 3*80 + 16), bit-identical to (float)(g%80)+0.5f.
// ---------------------------------------------------------------------------
__global__ __launch_bounds__(256) void topk_kernel(const float* __restrict__ cxys,
                                                   int* __restrict__ topk) {
  const int t  = threadIdx.x;
  const int bn = blockIdx.x;                 // b*N_ + n
  const float cx = cxys[bn * 2 + 0];
  const float cy = cxys[bn * 2 + 1];

  float gx = (float)(t % GS_) + 0.5f;        // exact
  float gy = (float)(t / GS_) + 0.5f;        // exact
  unsigned int g = (unsigned int)t;

  unsigned long long k0 = ~0ull, k1 = ~0ull, k2 = ~0ull;
  for (int i = 0; i < NG_ / 256; ++i) {
    const float d = fabsf(gx - cx) + fabsf(gy - cy);
    const unsigned long long key =
        ((unsigned long long)__float_as_uint(d) << 32) | g;
    if (key < k0)      { k2 = k1; k1 = k0; k0 = key; }
    else if (key < k1) { k2 = k1; k1 = key; }
    else if (key < k2) { k2 = key; }
    // advance by 256 cells: x += 16 (mod 80), y += 3 or 4 (all exact in f32)
    gx += 16.0f;
    if (gx >= 80.5f) { gx -= 80.0f; gy += 4.0f; }
    else             { gy += 3.0f; }
    g += 256u;
  }

  __shared__ unsigned long long sk[256 * 3];
  sk[t*3+0] = k0; sk[t*3+1] = k1; sk[t*3+2] = k2;
  __syncthreads();
  // tree-merge sorted triples (wave32-safe: plain LDS reduction)
  for (int s = 128; s > 0; s >>= 1) {
    if (t < s) {
      unsigned long long a0 = sk[t*3+0], a1 = sk[t*3+1], a2 = sk[t*3+2];
      unsigned long long b0 = sk[(t+s)*3+0], b1 = sk[(t+s)*3+1], b2 = sk[(t+s)*3+2];
      unsigned long long r0, r1, r2;
      if (a0 <= b0) {
        r0 = a0;
        if (a1 <= b0) { r1 = a1; r2 = (a2 <= b0) ? a2 : b0; }
        else          { r1 = b0; r2 = (a1 <= b1) ? a1 : b1; }
      } else {
        r0 = b0;
        if (b1 <= a0) { r1 = b1; r2 = (b2 <= a0) ? b2 : a0; }
        else          { r1 = a0; r2 = (a1 <= b1) ? a1 : b1; }
      }
      sk[t*3+0] = r0; sk[t*3+1] = r1; sk[t*3+2] = r2;
    }
    __syncthreads();
  }
  if (t < TOPK_) topk[bn * TOPK_ + t] = (int)(sk[t] & 0xffffffffull);
}

// ---------------------------------------------------------------------------
// Kernel B: per (b,g) resolve assignments.
//   pos bitmask over n: g in topk[b,n] AND mask_gt[b,n,g] != 0
//   cnt==0 -> fg=0, tgt=0 (argmax of zeros)
//   cnt==1 -> fg=1, tgt=ctz(pos)
//   cnt>1  -> nearest GT (argmin_n dist, first-index ties) keeps its mask bit
// GT metadata staged into LDS with CDNA5 async global->LDS loads.
// ---------------------------------------------------------------------------
__global__ __launch_bounds__(256) void assign_kernel(
    const int* __restrict__ topk, const float* __restrict__ cxys,
    const float* __restrict__ mask_gt, int* __restrict__ tgt,
    float* __restrict__ fg) {
  __shared__ int   s_tk[N_ * TOPK_];   // 192 ints
  __shared__ float s_cxy[N_ * 2];      // 128 floats (x,y interleaved)
  const int t = threadIdx.x;
  const int b = blockIdx.y;

  if (t < N_ * TOPK_) async_g2l_b32(&topk[b * N_ * TOPK_ + t], &s_tk[t]);
  if (t < N_ * 2)     async_g2l_b32(&cxys[b * N_ * 2 + t], &s_cxy[t]);
  wait_async_lds();
  __syncthreads();

  const int g = blockIdx.x * 256 + t;
  const float* mrow = mask_gt + (size_t)b * N_ * NG_ + g;

  unsigned long long pos = 0ull;
  for (int n = 0; n < N_; ++n) {
    const int o = n * 3;
    if (g == s_tk[o] || g == s_tk[o + 1] || g == s_tk[o + 2]) {
      if (mrow[(size_t)n * NG_] != 0.0f) pos |= (1ull << n);
    }
  }
  const int cnt = __builtin_popcountll(pos);

  // argmin over all n (first index wins on ties -> strict <)
  const float gx = (float)(g % GS_) + 0.5f;
  const float gy = (float)(g / GS_) + 0.5f;
  float best = 3.402823466e38f;
  int bi = 0;
  for (int n = 0; n < N_; ++n) {
    const float d = fabsf(gx - s_cxy[2*n]) + fabsf(gy - s_cxy[2*n + 1]);
    if (d < best) { best = d; bi = n; }
  }

  int tg; float f;
  if (cnt == 0)      { tg = 0; f = 0.0f; }
  else if (cnt == 1) { tg = __builtin_ctzll(pos); f = 1.0f; }
  else {
    const bool v = (pos >> bi) & 1ull;
    f  = v ? 1.0f : 0.0f;
    tg = v ? bi : 0;
  }
  tgt[b * NG_ + g] = tg;
  fg [b * NG_ + g] = f;
}

// ---------------------------------------------------------------------------
// Kernel C: bboxes (NT float4), fg_out with anchor-ratio filter, anc copy.
// ---------------------------------------------------------------------------
__global__ __launch_bounds__(256) void bbox_kernel(
    const float* __restrict__ anc_wh, const float* __restrict__ cxys,
    const float* __restrict__ whs, const int* __restrict__ tgt,
    const float* __restrict__ fg, float* __restrict__ out) {
  const int t = threadIdx.x;
  const int b = blockIdx.y;
  const int g = blockIdx.x * 256 + t;
  const int tg = tgt[b * NG_ + g];
  const float f = fg[b * NG_ + g];
  const float cx = cxys[(b * N_ + tg) * 2 + 0];
  const float cy = cxys[(b * N_ + tg) * 2 + 1];
  const float w  = whs [(b * N_ + tg) * 2 + 0];
  const float h  = whs [(b * N_ + tg) * 2 + 1];
  vfloat4 bb;
  bb.x = cx - (float)(g % GS_);              // txy = cxy - grid
  bb.y = cy - (float)(g / GS_);
  bb.z = w;
  bb.w = h;
  for (int a = 0; a < NA_; ++a) {
    const size_t base = ((size_t)(b * NA_ + a)) * NG_ + g;
    __builtin_nontemporal_store(bb, reinterpret_cast<vfloat4*>(out) + base);
    const float aw = anc_wh[2*a + 0], ah = anc_wh[2*a + 1];
    const float rw = w / aw, rh = h / ah;
    const float m  = fmaxf(fmaxf(rw, 1.0f / rw), fmaxf(rh, 1.0f / rh));
    const float fo = (f != 0.0f && m < 4.0f) ? 1.0f : 0.0f;
    __builtin_nontemporal_store(fo, out + FG_OFF + base);
  }
  if (b == 0 && blockIdx.x == 0 && t < 6) out[ANC_OFF + t] = anc_wh[t];
}

// ---------------------------------------------------------------------------
// Kernel D: one-hot scores, NT float4 per thread, fully coalesced streaming
// stores (196MB > 192MB L2 -> non-temporal avoids cache thrash).
// ---------------------------------------------------------------------------
__global__ __launch_bounds__(256) void score_kernel(
    const int* __restrict__ labels, const int* __restrict__ tgt,
    float* __restrict__ out_sc) {
  const int idx4 = blockIdx.x * 256 + threadIdx.x;     // < 12,288,000
  const int bag  = idx4 / (NC_ / 4);                   // (b*NA+a)*NG + g
  const int q    = idx4 - bag * (NC_ / 4);
  const int bA   = bag / NG_;
  const int g    = bag - bA * NG_;
  const int b    = bA / NA_;
  const int tg   = tgt[b * NG_ + g];
  const int lab  = labels[b * N_ + tg];
  const int c0   = q * 4;
  vfloat4 v;
  v.x = (lab == c0    ) ? 1.0f : 0.0f;
  v.y = (lab == c0 + 1) ? 1.0f : 0.0f;
  v.z = (lab == c0 + 2) ? 1.0f : 0.0f;
  v.w = (lab == c0 + 3) ? 1.0f : 0.0f;
  __builtin_nontemporal_store(v, reinterpret_cast<vfloat4*>(out_sc) + idx4);
}

// ---------------------------------------------------------------------------
extern "C" void kernel_launch(void* const* d_in, const int* in_sizes, int n_in,
                              void* d_out, int out_size, void* d_ws, size_t ws_size,
                              hipStream_t stream) {
  (void)in_sizes; (void)n_in; (void)out_size; (void)ws_size;
  const float* anc    = (const float*)d_in[0];
  // d_in[1] = grid: recomputed exactly on device as (g%80, g/80)
  const int*   labels = (const int*)d_in[2];
  const float* cxys   = (const float*)d_in[3];
  const float* whs    = (const float*)d_in[4];
  const float* maskgt = (const float*)d_in[5];
  float* out = (float*)d_out;

  char* ws = (char*)d_ws;                         // needs ~1.8 MB
  int*   topk = (int*)ws;                                  // B*N*3 ints
  int*   tgt  = (int*)(ws + 64 * 1024);                    // B*NG ints
  float* fgm  = (float*)(ws + 64 * 1024 + 896 * 1024);     // B*NG floats

  topk_kernel  <<<B_ * N_,              256, 0, stream>>>(cxys, topk);
  assign_kernel<<<dim3(NG_/256, B_),    256, 0, stream>>>(topk, cxys, maskgt, tgt, fgm);
  bbox_kernel  <<<dim3(NG_/256, B_),    256, 0, stream>>>(anc, cxys, whs, tgt, fgm, out);
  score_kernel <<<(B_*NA_*NG_*(NC_/4))/256, 256, 0, stream>>>(labels, tgt, out + SC_OFF);
}